// LSTMClassifier_89936615178378
// MI455X (gfx1250) — compile-verified
//
#include <hip/hip_runtime.h>
#include <math.h>
#include <stdint.h>

typedef __attribute__((ext_vector_type(2))) float v2f;
typedef __attribute__((ext_vector_type(8))) float v8f;

#define BB 64
#define TT 512
#define EE 512
#define HH 1024
#define OO 5

// chunking of the K dimension for LDS staging
#define KCHUNK     64
#define NCHUNK_IN  (EE / KCHUNK)             // 8
#define NCHUNK_HID (HH / KCHUNK)             // 16
#define NCHUNK_TOT (NCHUNK_IN + NCHUNK_HID)  // 24

// ---------------------------------------------------------------------------
// gfx1250 async copy global->LDS (ASYNCcnt-tracked, bypasses VGPRs).
// lds_off: byte offset in LDS; gaddr: 64-bit global address. 16B per lane.
// ---------------------------------------------------------------------------
__device__ __forceinline__ void async_copy_b128(uint32_t lds_off, uint64_t gaddr) {
    asm volatile("global_load_async_to_lds_b128 %0, %1, off"
                 :: "v"(lds_off), "v"(gaddr)
                 : "memory");
}
__device__ __forceinline__ void wait_asynccnt0() {
    asm volatile("s_wait_asynccnt 0x0" ::: "memory");
}

// ---------------------------------------------------------------------------
// Zero h/c double buffers + final-hidden buffer each call (ws is poisoned).
// ---------------------------------------------------------------------------
__global__ void init_ws_kernel(float* __restrict__ ws, int n) {
    int i = blockIdx.x * blockDim.x + threadIdx.x;
    if (i < n) ws[i] = 0.0f;
}

// ---------------------------------------------------------------------------
// lengths[b] = count of non-pad tokens (x != 1) in row b.
// ---------------------------------------------------------------------------
__global__ void lengths_kernel(const int* __restrict__ x, int* __restrict__ lengths) {
    int b = threadIdx.x;
    if (b < BB) {
        int cnt = 0;
        for (int t = 0; t < TT; ++t) cnt += (x[b * TT + t] != 1) ? 1 : 0;
        lengths[b] = cnt;
    }
}

__device__ __forceinline__ float sigmoidf_(float v) {
    return 1.0f / (1.0f + expf(-v));
}

// ---------------------------------------------------------------------------
// One LSTM timestep.
// Grid: HH/16 = 64 blocks (one 16-column strip of H each).
// Block: 128 threads = 4 wave32s; wave w owns batch-row tile w (rows 16w..16w+15).
//
// Weights for the block's 16 columns are staged chunk-by-chunk (64 K-rows,
// 3 gates) into a double-buffered LDS tile with global_load_async_to_lds_b128,
// then consumed by fp32 WMMA (16x16x4). Chunk c+1 streams while chunk c
// computes; s_wait_asynccnt 0 + s_barrier publish each buffer.
//
// WMMA f32 per-lane layouts:
//   A (16x4):  lane L -> row M=L&15, K=(L>>4)*2 and +1 in its two VGPRs.
//   B (4x16):  lane L -> col N=L&15, K=(L>>4)*2 and +1.
//   C/D (16x16, 8 VGPRs): VGPR r, lane L -> M = r + 8*(L>>4), N = L&15.
// ---------------------------------------------------------------------------
__global__ __launch_bounds__(128)
void lstm_step_kernel(int t,
                      const int*   __restrict__ x,
                      const float* __restrict__ embed_w,
                      const float* __restrict__ g_win, const float* __restrict__ g_whid, const float* __restrict__ g_b,
                      const float* __restrict__ i_win, const float* __restrict__ i_whid, const float* __restrict__ i_b,
                      const float* __restrict__ o_win, const float* __restrict__ o_whid, const float* __restrict__ o_b,
                      const float* __restrict__ h_in,  const float* __restrict__ c_in,
                      float* __restrict__ h_out, float* __restrict__ c_out,
                      const int* __restrict__ lengths, float* __restrict__ final_h)
{
    // double-buffered weight tiles: [2][3 gates][KCHUNK rows][16 cols]
    __shared__ float smem[2 * 3 * KCHUNK * 16];

    const int tid   = threadIdx.x;        // 0..127
    const int lane  = tid & 31;
    const int wave  = tid >> 5;           // 0..3  -> M tile (batch rows)
    const int tileN = blockIdx.x;         // 0..63 -> 16-column strip of H
    const int n0    = tileN * 16;
    const int m     = lane & 15;
    const int kp    = (lane >> 4) << 1;   // 0 or 2
    const int n     = n0 + (lane & 15);

    // A-operand row for this lane (batch row), embedding gather fused in.
    const int   arow = wave * 16 + m;
    const int   tok  = x[arow * TT + t];
    const float* __restrict__ emb_row = embed_w + (size_t)tok * EE;
    const float* __restrict__ h_row   = h_in + (size_t)arow * HH;

    const uint32_t lds_base = (uint32_t)(uintptr_t)smem;

    // staging geometry: 24 async-B128 per chunk (3 gates * 64 rows * 64B),
    // 6 per wave; each instr copies 8 rows (4 lanes of 16B per row).
    const int st_row_in_seg = lane >> 2;          // 0..7
    const int st_col_byte   = (lane & 3) << 4;    // 0,16,32,48

    // helper lambda: stage chunk `c` (global chunk index) into LDS buffer `d`
    auto stage_chunk = [&](int c, int d) {
        const float* Wg;
        const float* Wi;
        const float* Wo;
        int k0;
        if (c < NCHUNK_IN) { Wg = g_win;  Wi = i_win;  Wo = o_win;  k0 = c * KCHUNK; }
        else               { Wg = g_whid; Wi = i_whid; Wo = o_whid; k0 = (c - NCHUNK_IN) * KCHUNK; }
        const float* Ws[3] = { Wg, Wi, Wo };
        #pragma unroll
        for (int j = 0; j < 6; ++j) {
            const int seg     = wave * 6 + j;       // 0..23
            const int gate    = seg >> 3;           // 0..2
            const int rowbase = (seg & 7) * 8;      // 0..56
            const int row     = rowbase + st_row_in_seg;
            const uint64_t ga = (uint64_t)(uintptr_t)(Ws[gate]
                               + (size_t)(k0 + row) * HH + n0) + (uint64_t)st_col_byte;
            const uint32_t la = lds_base
                               + (uint32_t)(((d * 3 + gate) * KCHUNK + row) * 64)
                               + (uint32_t)st_col_byte;
            async_copy_b128(la, ga);
        }
    };

    v8f accg = {}; v8f acci = {}; v8f acco = {};

    // prologue: stage chunk 0 into buffer 0
    stage_chunk(0, 0);
    wait_asynccnt0();
    __syncthreads();

    for (int c = 0; c < NCHUNK_TOT; ++c) {
        const int d = c & 1;
        // stream next chunk into the other buffer
        if (c + 1 < NCHUNK_TOT) stage_chunk(c + 1, d ^ 1);

        const float* Arow;
        int k0;
        if (c < NCHUNK_IN) { Arow = emb_row; k0 = c * KCHUNK; }
        else               { Arow = h_row;   k0 = (c - NCHUNK_IN) * KCHUNK; }

        const float* __restrict__ sg = smem + (d * 3 + 0) * KCHUNK * 16;
        const float* __restrict__ si = smem + (d * 3 + 1) * KCHUNK * 16;
        const float* __restrict__ so = smem + (d * 3 + 2) * KCHUNK * 16;

        #pragma unroll 4
        for (int kk = 0; kk < KCHUNK; kk += 4) {
            v2f a;
            a.x = Arow[k0 + kk + kp];
            a.y = Arow[k0 + kk + kp + 1];
            const int r0 = (kk + kp) * 16 + (lane & 15);
            v2f bg, bi, bo;
            bg.x = sg[r0];  bg.y = sg[r0 + 16];
            bi.x = si[r0];  bi.y = si[r0 + 16];
            bo.x = so[r0];  bo.y = so[r0 + 16];
            accg = __builtin_amdgcn_wmma_f32_16x16x4_f32(false, a, false, bg, (short)0, accg, false, false);
            acci = __builtin_amdgcn_wmma_f32_16x16x4_f32(false, a, false, bi, (short)0, acci, false, false);
            acco = __builtin_amdgcn_wmma_f32_16x16x4_f32(false, a, false, bo, (short)0, acco, false, false);
        }

        // publish the streamed buffer; also guards re-staging of buffer d
        wait_asynccnt0();
        __syncthreads();
    }

    // ---- elementwise cell update (C/D layout: M = r + 8*(lane>>4), N = lane&15) ----
    const int   hl  = lane >> 4;
    const float bgc = g_b[n];
    const float bic = i_b[n];
    const float boc = o_b[n];

    #pragma unroll
    for (int r = 0; r < 8; ++r) {
        const int brow = wave * 16 + r + 8 * hl;
        const float pg = accg[r] + bgc;
        const float pi = acci[r] + bic;
        const float po = acco[r] + boc;

        const float g  = tanhf(pg);
        const float ig = sigmoidf_(pi);          // f gate == i gate (source bug)
        const float og = sigmoidf_(po);

        const float cprev = c_in[(size_t)brow * HH + n];
        const float cnew  = g * ig + cprev * ig;
        const float hnew  = tanhf(cnew) * og;

        c_out[(size_t)brow * HH + n] = cnew;
        h_out[(size_t)brow * HH + n] = hnew;

        // Capture hidden state at index lengths[b]-1, zeroed if that position
        // is itself a pad token (matches reference's outs masking + gather).
        if (t == lengths[brow] - 1) {
            const bool is_pad = (x[brow * TT + t] == 1);
            final_h[(size_t)brow * HH + n] = is_pad ? 0.0f : hnew;
        }
    }
}

// ---------------------------------------------------------------------------
// logits[b,o] = dot(final_h[b,:], out_w[o,:]) + out_b[o]
// Grid: B*O blocks, 256 threads (8 wave32s), shuffle + LDS reduction.
// ---------------------------------------------------------------------------
__global__ __launch_bounds__(256)
void proj_kernel(const float* __restrict__ final_h,
                 const float* __restrict__ out_w,
                 const float* __restrict__ out_b,
                 float* __restrict__ logits)
{
    __shared__ float red[8];
    const int b = blockIdx.x / OO;
    const int o = blockIdx.x % OO;

    float s = 0.0f;
    for (int h = threadIdx.x; h < HH; h += 256)
        s += final_h[(size_t)b * HH + h] * out_w[(size_t)o * HH + h];

    // wave32 reduction
    #pragma unroll
    for (int off = 16; off > 0; off >>= 1)
        s += __shfl_down(s, off, 32);

    if ((threadIdx.x & 31) == 0) red[threadIdx.x >> 5] = s;
    __syncthreads();
    if (threadIdx.x == 0) {
        float tot = 0.0f;
        #pragma unroll
        for (int w = 0; w < 8; ++w) tot += red[w];
        logits[b * OO + o] = tot + out_b[o];
    }
}

// ---------------------------------------------------------------------------
extern "C" void kernel_launch(void* const* d_in, const int* in_sizes, int n_in,
                              void* d_out, int out_size, void* d_ws, size_t ws_size,
                              hipStream_t stream) {
    const int*   x       = (const int*)  d_in[0];
    const float* embed_w = (const float*)d_in[1];
    const float* g_win   = (const float*)d_in[2];
    const float* g_whid  = (const float*)d_in[3];
    const float* g_b     = (const float*)d_in[4];
    const float* i_win   = (const float*)d_in[5];
    const float* i_whid  = (const float*)d_in[6];
    const float* i_b     = (const float*)d_in[7];
    // d_in[8..10] = f_* params: unused (reference bug makes f == i)
    const float* o_win   = (const float*)d_in[11];
    const float* o_whid  = (const float*)d_in[12];
    const float* o_b     = (const float*)d_in[13];
    const float* out_w   = (const float*)d_in[14];
    const float* out_b   = (const float*)d_in[15];
    float* logits = (float*)d_out;

    // Workspace layout (floats): h0, h1, c0, c1, final, then lengths (ints).
    float* ws = (float*)d_ws;
    const int BH = BB * HH;
    float* hbuf[2]  = { ws,          ws + BH     };
    float* cbuf[2]  = { ws + 2 * BH, ws + 3 * BH };
    float* final_h  = ws + 4 * BH;
    int*   lengths  = (int*)(ws + 5 * BH);

    const int zn = 5 * BH;
    init_ws_kernel<<<(zn + 255) / 256, 256, 0, stream>>>(ws, zn);
    lengths_kernel<<<1, 64, 0, stream>>>(x, lengths);

    for (int t = 0; t < TT; ++t) {
        lstm_step_kernel<<<HH / 16, 128, 0, stream>>>(
            t, x, embed_w,
            g_win, g_whid, g_b,
            i_win, i_whid, i_b,
            o_win, o_whid, o_b,
            hbuf[t & 1], cbuf[t & 1],
            hbuf[(t + 1) & 1], cbuf[(t + 1) & 1],
            lengths, final_h);
    }

    proj_kernel<<<BB * OO, 256, 0, stream>>>(final_h, out_w, out_b, logits);
}